// SelfSupPredictor_55808805044924
// MI455X (gfx1250) — compile-verified
//
#include <hip/hip_runtime.h>
#include <hip/hip_bf16.h>

// ---------------------------------------------------------------------------
// SelfSupPredictor on MI455X (gfx1250, wave32).
// fp32 end-to-end; matrix ops use V_WMMA_F32_16X16X4_F32 (full fp32 WMMA).
// Edge aggregation via once-per-launch CSR build (no float atomics in the 11
// message-passing passes), fused with the WMMA linear layer through LDS.
// ---------------------------------------------------------------------------

typedef __attribute__((ext_vector_type(2))) float v2f;
typedef __attribute__((ext_vector_type(8))) float v8f;

#define Nn   100000
#define Ee   1600000
#define Ff   64
#define Vv   2
#define Aa   6
#define Pp   8192
#define MT_N (Nn / 16)   // 6250 row tiles for node GEMMs
#define MT_P (Pp / 16)   // 512  row tiles for decode GEMM
#define NB   ((Nn + 255) / 256)   // 391 scan blocks

// ---------------- utility kernels ----------------

__global__ void k_zero(float* __restrict__ buf, int n) {
    int i = blockIdx.x * blockDim.x + threadIdx.x;
    if (i < n) buf[i] = 0.f;
}

__global__ void k_izero(int* __restrict__ buf, int n) {
    int i = blockIdx.x * blockDim.x + threadIdx.x;
    if (i < n) buf[i] = 0;
}

__global__ void k_icopy(int* __restrict__ dst, const int* __restrict__ src, int n) {
    int i = blockIdx.x * blockDim.x + threadIdx.x;
    if (i < n) dst[i] = src[i];
}

__global__ void k_add_into(float* __restrict__ dst, const float* __restrict__ src, int n) {
    int i = blockIdx.x * blockDim.x + threadIdx.x;
    if (i < n) dst[i] += src[i];
}

// ---------------- CSR build (once per launch; edge list is static) ----------

__global__ void k_hist(const int* __restrict__ edst, int* __restrict__ deg, int ne) {
    int i = blockIdx.x * blockDim.x + threadIdx.x;
    if (i < ne) atomicAdd(&deg[edst[i]], 1);
}

// phase A: per-256-block sums of deg
__global__ void k_blocksum(const int* __restrict__ deg, int* __restrict__ part, int n) {
    __shared__ int sd[256];
    int i = blockIdx.x * 256 + threadIdx.x;
    sd[threadIdx.x] = (i < n) ? deg[i] : 0;
    __syncthreads();
    for (int off = 128; off > 0; off >>= 1) {
        if ((int)threadIdx.x < off) sd[threadIdx.x] += sd[threadIdx.x + off];
        __syncthreads();
    }
    if (threadIdx.x == 0) part[blockIdx.x] = sd[0];
}

// phase B: single-block exclusive scan of the NB partials; also rowptr[N]=E
__global__ void k_scan_part(int* __restrict__ part, int nb, int* __restrict__ rowptr_end) {
    __shared__ int sh[512];
    int t = threadIdx.x;
    int v = (t < nb) ? part[t] : 0;
    sh[t] = v;
    __syncthreads();
    for (int off = 1; off < 512; off <<= 1) {
        int u = (t >= off) ? sh[t - off] : 0;
        __syncthreads();
        sh[t] += u;
        __syncthreads();
    }
    if (t < nb) part[t] = sh[t] - v;   // exclusive
    if (t == 511) rowptr_end[0] = sh[511];
}

// phase C: in-block exclusive scan + partial offset -> rowptr
__global__ void k_scan_final(const int* __restrict__ deg, const int* __restrict__ part,
                             int* __restrict__ rowptr, int n) {
    __shared__ int sh[256];
    int i = blockIdx.x * 256 + threadIdx.x;
    int v = (i < n) ? deg[i] : 0;
    sh[threadIdx.x] = v;
    __syncthreads();
    for (int off = 1; off < 256; off <<= 1) {
        int u = ((int)threadIdx.x >= off) ? sh[threadIdx.x - off] : 0;
        __syncthreads();
        sh[threadIdx.x] += u;
        __syncthreads();
    }
    if (i < n) rowptr[i] = part[blockIdx.x] + sh[threadIdx.x] - v;
}

__global__ void k_scatter_edges(const int* __restrict__ esrc, const int* __restrict__ edst,
                                int* __restrict__ cursor, int* __restrict__ srt, int ne) {
    int i = blockIdx.x * blockDim.x + threadIdx.x;
    if (i < ne) {
        int pos = atomicAdd(&cursor[edst[i]], 1);
        srt[pos] = esrc[i];
    }
}

// insertion-sort each row's source list -> deterministic aggregation order
__global__ void k_sort_rows(const int* __restrict__ rowptr, int* __restrict__ srt, int n) {
    int i = blockIdx.x * blockDim.x + threadIdx.x;
    if (i >= n) return;
    int a = rowptr[i], b = rowptr[i + 1];
    for (int j = a + 1; j < b; ++j) {
        int key = srt[j];
        int k = j - 1;
        while (k >= a && srt[k] > key) { srt[k + 1] = srt[k]; --k; }
        srt[k + 1] = key;
    }
}

// ---------------- fused GCN layer: out = relu(csr_agg(h)@W + h@Wself + b) ----
// 128-thread block = one 16-node M-tile. Phase 1: 4 waves aggregate 4 nodes
// each into LDS (gather-only, no atomics) and stage the 16 self rows in LDS.
// Phase 2: each wave computes one 16x16 output tile with fp32 WMMA (K=64).

__global__ void __launch_bounds__(128)
k_gcn_fused(const float* __restrict__ h,
            const int* __restrict__ rowptr,
            const int* __restrict__ srt,
            const float* __restrict__ W,
            const float* __restrict__ Wself,
            const float* __restrict__ bias,
            float* __restrict__ out) {
    __shared__ float sA[16 * Ff];   // aggregated rows
    __shared__ float sH[16 * Ff];   // self rows
    int w    = threadIdx.x >> 5;    // wave 0..3
    int lane = threadIdx.x & 31;
    int n0   = blockIdx.x * 16;

    for (int i = 0; i < 4; ++i) {
        int li = w * 4 + i;         // local node 0..15
        int n  = n0 + li;
        int start = rowptr[n], end = rowptr[n + 1];
        float ax = 0.f, ay = 0.f;
        for (int base = start; base < end; base += 32) {
            int rem   = end - base;
            int cnt   = rem < 32 ? rem : 32;
            int myidx = (lane < rem) ? srt[base + lane] : 0;
            __builtin_prefetch(srt + base + 32 + lane, 0, 0);   // global_prefetch_b8
            for (int j = 0; j < cnt; ++j) {
                int s = __shfl(myidx, j, 32);
                const float* r = h + (size_t)s * Ff + 2 * lane;
                ax += r[0];
                ay += r[1];
            }
        }
        sA[li * Ff + 2 * lane]     = ax;
        sA[li * Ff + 2 * lane + 1] = ay;
        const float* hr = h + (size_t)n * Ff + 2 * lane;   // stage self row (coalesced)
        sH[li * Ff + 2 * lane]     = hr[0];
        sH[li * Ff + 2 * lane + 1] = hr[1];
    }
    __syncthreads();

    // WMMA phase: wave w -> column tile w
    int hf  = lane >> 4;
    int l   = lane & 15;
    int col = w * 16 + l;
    const float* arow = sA + l * Ff;
    const float* hrow = sH + l * Ff;

    v8f c = {};
#pragma unroll
    for (int k0 = 0; k0 < Ff; k0 += 4) {
        int ka = k0 + 2 * hf;
        v2f aA; aA.x = arow[ka];             aA.y = arow[ka + 1];
        v2f aH; aH.x = hrow[ka];             aH.y = hrow[ka + 1];
        v2f bW; bW.x = W[ka * Ff + col];     bW.y = W[(ka + 1) * Ff + col];
        v2f bS; bS.x = Wself[ka * Ff + col]; bS.y = Wself[(ka + 1) * Ff + col];
        c = __builtin_amdgcn_wmma_f32_16x16x4_f32(false, aA, false, bW, (short)0, c, false, false);
        c = __builtin_amdgcn_wmma_f32_16x16x4_f32(false, aH, false, bS, (short)0, c, false, false);
    }

    float bc = bias[col];
#pragma unroll
    for (int r = 0; r < 8; ++r) {
        int orow = n0 + r + 8 * hf;   // C layout: VGPR r -> rows r / r+8
        float v = c[r] + bc;
        out[(size_t)orow * Ff + col] = v > 0.f ? v : 0.f;
    }
}

// ---------------- decode MLP layer 1: h = relu([f[p],f[s],f[d]] @ Wd1 + bd1) --

__global__ void k_decode_mlp1(const float* __restrict__ feat,
                              const int* __restrict__ p,
                              const int* __restrict__ s,
                              const int* __restrict__ d,
                              const float* __restrict__ Wd1,
                              const float* __restrict__ bd1,
                              float* __restrict__ hmid) {
    int wid  = blockIdx.x * (blockDim.x >> 5) + (threadIdx.x >> 5);
    int lane = threadIdx.x & 31;
    int mt = wid >> 2;
    int nt = wid & 3;
    if (mt >= MT_P) return;

    int hf = lane >> 4;
    int l  = lane & 15;
    int t  = mt * 16 + l;       // triple index = A row
    const float* b0 = feat + (size_t)p[t] * Ff;
    const float* b1 = feat + (size_t)s[t] * Ff;
    const float* b2 = feat + (size_t)d[t] * Ff;
    int col = nt * 16 + l;

    v8f c = {};
#pragma unroll
    for (int k0 = 0; k0 < 3 * Ff; k0 += 4) {
        int ka = k0 + 2 * hf;                       // even; ka,ka+1 stay in one 64-block
        const float* bp = (ka < Ff) ? b0 : (ka < 2 * Ff) ? b1 : b2;
        int kk = ka & (Ff - 1);
        v2f a; a.x = bp[kk];              a.y = bp[kk + 1];
        v2f b; b.x = Wd1[ka * Ff + col];  b.y = Wd1[(ka + 1) * Ff + col];
        c = __builtin_amdgcn_wmma_f32_16x16x4_f32(false, a, false, b, (short)0, c, false, false);
    }

    float bc = bd1[col];
#pragma unroll
    for (int r = 0; r < 8; ++r) {
        int orow = mt * 16 + r + 8 * hf;
        float v = c[r] + bc;
        hmid[(size_t)orow * Ff + col] = v > 0.f ? v : 0.f;
    }
}

// ---------------- decode MLP layer 2 + log_sigmoid + scatter into pred -------

__global__ void k_decode_mlp2(const float* __restrict__ hmid,
                              const float* __restrict__ Wd2,
                              const float* __restrict__ bd2,
                              const int* __restrict__ p,
                              float* __restrict__ pred) {
    int t = blockIdx.x * blockDim.x + threadIdx.x;
    if (t >= Pp) return;
    float z = bd2[0];
    const float* hr = hmid + (size_t)t * Ff;
#pragma unroll 8
    for (int k = 0; k < Ff; ++k) z += hr[k] * Wd2[k];
    // numerically stable log_sigmoid
    float ls = (z >= 0.f) ? -log1pf(expf(-z)) : (z - log1pf(expf(z)));
    atomicAdd(&pred[p[t]], ls);
}

// ---------------- prev_prob chain ----------------

__global__ void k_gather_sum(const float* __restrict__ pred,
                             const int* __restrict__ p,
                             float* __restrict__ tmp) {
    __shared__ float sd[256];
    int i = blockIdx.x * blockDim.x + threadIdx.x;
    float v = (i < Pp) ? pred[p[i]] : 0.f;
    sd[threadIdx.x] = v;
    __syncthreads();
    for (int off = 128; off > 0; off >>= 1) {
        if ((int)threadIdx.x < off) sd[threadIdx.x] += sd[threadIdx.x + off];
        __syncthreads();
    }
    if (threadIdx.x == 0) atomicAdd(tmp, sd[0]);
}

__global__ void k_bump(float* __restrict__ prevp, const float* __restrict__ tmp) {
    prevp[0] += tmp[0];
}

__global__ void k_scatter_prev(float* __restrict__ pred,
                               const int* __restrict__ p,
                               const float* __restrict__ prevp) {
    int t = blockIdx.x * blockDim.x + threadIdx.x;
    if (t >= Pp) return;
    atomicAdd(&pred[p[t]], prevp[0]);
}

// ---------------- host-side orchestration ----------------

extern "C" void kernel_launch(void* const* d_in, const int* in_sizes, int n_in,
                              void* d_out, int out_size, void* d_ws, size_t ws_size,
                              hipStream_t stream) {
    const float* x      = (const float*)d_in[0];
    const int*   eidx   = (const int*)  d_in[1];
    const int*   places = (const int*)  d_in[2];
    const int*   srcs   = (const int*)  d_in[3];
    const int*   dsts   = (const int*)  d_in[4];
    const float* W1     = (const float*)d_in[5];
    const float* Ws1    = (const float*)d_in[6];
    const float* b1     = (const float*)d_in[7];
    const float* W2     = (const float*)d_in[8];
    const float* Ws2    = (const float*)d_in[9];
    const float* b2     = (const float*)d_in[10];
    const float* Wd1    = (const float*)d_in[11];
    const float* bd1    = (const float*)d_in[12];
    const float* Wd2    = (const float*)d_in[13];
    const float* bd2    = (const float*)d_in[14];
    float* out = (float*)d_out;

    const int* esrc = eidx;        // edge_index[0]
    const int* edst = eidx + Ee;   // edge_index[1]

    // workspace layout: floats first, then ints
    const size_t NF = (size_t)Nn * Ff;
    float* feat0 = (float*)d_ws;
    float* featA = feat0 + NF;
    float* featB = featA + NF;
    float* hmid  = featB + NF;                // Pp*Ff
    float* pred  = hmid + (size_t)Pp * Ff;    // Nn
    float* tmp   = pred + Nn;
    float* prevp = tmp + 1;
    int*   deg    = (int*)(prevp + 1);
    int*   rowptr = deg + Nn;                 // Nn+1
    int*   cursor = rowptr + Nn + 1;          // Nn
    int*   part   = cursor + Nn;              // 512
    int*   srt    = part + 512;               // Ee

    const int B = 256;
    const int gN    = (Nn + B - 1) / B;
    const int gE    = (Ee + B - 1) / B;
    const int gDec1 = (MT_P * 4 + 7) / 8;
    const int gP    = (Pp + B - 1) / B;

    // ---- build CSR (amortized over all 11 message-passing passes) ----
    k_izero<<<gN, B, 0, stream>>>(deg, Nn);
    k_hist<<<gE, B, 0, stream>>>(edst, deg, Ee);
    k_blocksum<<<NB, B, 0, stream>>>(deg, part, Nn);
    k_scan_part<<<1, 512, 0, stream>>>(part, NB, rowptr + Nn);
    k_scan_final<<<NB, B, 0, stream>>>(deg, part, rowptr, Nn);
    k_icopy<<<gN, B, 0, stream>>>(cursor, rowptr, Nn);
    k_scatter_edges<<<gE, B, 0, stream>>>(esrc, edst, cursor, srt, Ee);
    k_sort_rows<<<gN, B, 0, stream>>>(rowptr, srt, Nn);

    // predictions = 0
    k_zero<<<gN, B, 0, stream>>>(out, Nn);

    // first encoder: feat0 = relu(csr_agg(x)@W1 + x@Ws1 + b1)
    k_gcn_fused<<<MT_N, 128, 0, stream>>>(x, rowptr, srt, W1, Ws1, b1, feat0);

    for (int v = 0; v < Vv; ++v) {
        k_zero<<<gN, B, 0, stream>>>(pred, Nn);
        k_zero<<<1, 32, 0, stream>>>(prevp, 1);

        const float* cur = feat0;   // feat resets to post-first-encoder per variant
        float* nxt = featA;
        for (int a = 0; a < Aa; ++a) {
            const int* pp = places + ((size_t)v * Aa + a) * Pp;
            const int* ss = srcs   + ((size_t)v * Aa + a) * Pp;
            const int* dd = dsts   + ((size_t)v * Aa + a) * Pp;

            // decode + scatter pp into pred
            k_decode_mlp1<<<gDec1, B, 0, stream>>>(cur, pp, ss, dd, Wd1, bd1, hmid);
            k_decode_mlp2<<<gP, B, 0, stream>>>(hmid, Wd2, bd2, pp, pred);

            // prev_prob += sum(pred[p]); pred[p] += prev_prob (per occurrence)
            k_zero<<<1, 32, 0, stream>>>(tmp, 1);
            k_gather_sum<<<gP, B, 0, stream>>>(pred, pp, tmp);
            k_bump<<<1, 1, 0, stream>>>(prevp, tmp);
            k_scatter_prev<<<gP, B, 0, stream>>>(pred, pp, prevp);

            // second encoder (result of last iteration is never read -> skip)
            if (a < Aa - 1) {
                k_gcn_fused<<<MT_N, 128, 0, stream>>>(cur, rowptr, srt, W2, Ws2, b2, nxt);
                cur = nxt;
                nxt = (cur == featA) ? featB : featA;
            }
        }
        k_add_into<<<gN, B, 0, stream>>>(out, pred, Nn);
    }
    (void)in_sizes; (void)n_in; (void)out_size; (void)ws_size;
}